// MultiGATv2_4870492914032
// MI455X (gfx1250) — compile-verified
//
#include <hip/hip_runtime.h>
#include <math.h>

typedef __attribute__((ext_vector_type(2))) float v2f;
typedef __attribute__((ext_vector_type(8))) float v8f;

#define D 128
#define NEG_ATT 0.2f
#define NEG_OUT 0.01f

// float atomic max via monotonic int-bit trick (requires init to -inf)
__device__ __forceinline__ void atomicMaxF(float* addr, float v) {
  if (v >= 0.0f) atomicMax((int*)addr, __float_as_int(v));
  else           atomicMin((unsigned int*)addr, __float_as_uint(v));
}

// ---------------------------------------------------------------------------
// Kernel 1: x_l = x@Wl + bl, x_r = x@Wr + br via V_WMMA_F32_16X16X4_F32.
// Block = 128 threads (4 waves). Each block: one 16-row panel of x (staged in
// LDS), each wave computes 32 output columns for BOTH Wl and Wr.
// A layout (16x4 f32): lane m = lane&15, K = (lane>>4)*2 + vgpr
// B layout (4x16 f32): K = (lane>>4)*2 + vgpr, N = lane&15
// C layout (16x16 f32): M = vgpr + (lane>>4)*8, N = lane&15
// ---------------------------------------------------------------------------
__global__ __launch_bounds__(128) void gat_gemm(
    const float* __restrict__ x,
    const float* __restrict__ Wl, const float* __restrict__ bl,
    const float* __restrict__ Wr, const float* __restrict__ br,
    float* __restrict__ xl, float* __restrict__ xr, int N)
{
  __shared__ float sx[16 * D];   // 8 KB x-tile
  const int tid  = threadIdx.x;
  const int row0 = blockIdx.x * 16;

  // coalesced stage: 16 rows x 32 float4 = 512 float4 slots
  const float4* xg = (const float4*)x;
  float4* s4 = (float4*)sx;
#pragma unroll
  for (int i = 0; i < 4; ++i) {
    int idx = tid + 128 * i;            // 0..511
    int r   = idx >> 5;                 // row in tile
    int g   = row0 + r;
    float4 val = make_float4(0.f, 0.f, 0.f, 0.f);
    if (g < N) val = xg[(size_t)g * (D / 4) + (idx & 31)];
    s4[idx] = val;
  }
  __syncthreads();

  const int lane = tid & 31;
  const int wave = tid >> 5;
  const int mrow = lane & 15;
  const int kh   = lane >> 4;           // 0 or 1
  const int n0   = wave * 32;           // 32 columns per wave (2 n-tiles)

  v8f acc[2][2] = {};                   // [matrix(Wl/Wr)][ntile]

  for (int k0 = 0; k0 < D; k0 += 4) {
    const int ka = k0 + kh * 2;
    v2f a = *(const v2f*)&sx[mrow * D + ka];   // ds_load_b64
#pragma unroll
    for (int t = 0; t < 2; ++t) {
      const int n = n0 + t * 16 + mrow;
      v2f b0, b1;
      b0.x = Wl[(size_t)ka * D + n];
      b0.y = Wl[(size_t)(ka + 1) * D + n];
      b1.x = Wr[(size_t)ka * D + n];
      b1.y = Wr[(size_t)(ka + 1) * D + n];
      acc[0][t] = __builtin_amdgcn_wmma_f32_16x16x4_f32(
          false, a, false, b0, (short)0, acc[0][t], false, false);
      acc[1][t] = __builtin_amdgcn_wmma_f32_16x16x4_f32(
          false, a, false, b1, (short)0, acc[1][t], false, false);
    }
  }

#pragma unroll
  for (int t = 0; t < 2; ++t) {
    const int n = n0 + t * 16 + mrow;
    const float bLn = bl[n];
    const float bRn = br[n];
#pragma unroll
    for (int r = 0; r < 8; ++r) {
      const int m = row0 + r + kh * 8;
      if (m < N) {
        xl[(size_t)m * D + n] = acc[0][t][r] + bLn;
        xr[(size_t)m * D + n] = acc[1][t][r] + bRn;
      }
    }
  }
}

// ---------------------------------------------------------------------------
// Kernel 2: initialize accumulators (deterministic per launch)
// ---------------------------------------------------------------------------
__global__ void gat_init(float* __restrict__ out, float* __restrict__ mx,
                         float* __restrict__ z, int N)
{
  int i = blockIdx.x * blockDim.x + threadIdx.x;
  if (i < N * D) out[i] = 0.0f;
  if (i < N) { mx[i] = -INFINITY; z[i] = 0.0f; }
}

// ---------------------------------------------------------------------------
// Kernel 3: per-edge attention logit + segment max. One wave per edge.
// e = att . leaky_relu(x_l[src] + x_r[dst], 0.2)
// ---------------------------------------------------------------------------
__global__ __launch_bounds__(256) void gat_logit(
    const int* __restrict__ ei, const float* __restrict__ xl,
    const float* __restrict__ xr, const float* __restrict__ att,
    float* __restrict__ elog, float* __restrict__ mx, int E, int ET)
{
  int wave = (blockIdx.x * blockDim.x + threadIdx.x) >> 5;
  int lane = threadIdx.x & 31;
  if (wave >= ET) return;
  int src, dst;
  if (wave < E) { src = ei[wave]; dst = ei[E + wave]; }
  else          { src = wave - E; dst = src; }           // self loop

  float4 a = ((const float4*)att)[lane];
  float4 l = ((const float4*)(xl + (size_t)src * D))[lane];
  float4 r = ((const float4*)(xr + (size_t)dst * D))[lane];
  float4 s = make_float4(l.x + r.x, l.y + r.y, l.z + r.z, l.w + r.w);
  s.x = s.x > 0.f ? s.x : NEG_ATT * s.x;
  s.y = s.y > 0.f ? s.y : NEG_ATT * s.y;
  s.z = s.z > 0.f ? s.z : NEG_ATT * s.z;
  s.w = s.w > 0.f ? s.w : NEG_ATT * s.w;
  float v = s.x * a.x + s.y * a.y + s.z * a.z + s.w * a.w;
#pragma unroll
  for (int off = 16; off; off >>= 1) v += __shfl_xor(v, off, 32);
  if (lane == 0) {
    elog[wave] = v;
    atomicMaxF(&mx[dst], v);
  }
}

// ---------------------------------------------------------------------------
// Kernel 4: p = exp(e - m[dst]); z[dst] += p. One thread per edge.
// ---------------------------------------------------------------------------
__global__ void gat_expsum(const int* __restrict__ ei,
                           const float* __restrict__ elog,
                           const float* __restrict__ mx,
                           float* __restrict__ p, float* __restrict__ z,
                           int E, int ET)
{
  int e = blockIdx.x * blockDim.x + threadIdx.x;
  if (e >= ET) return;
  int dst = (e < E) ? ei[E + e] : (e - E);
  float pv = expf(elog[e] - mx[dst]);
  p[e] = pv;
  atomicAdd(&z[dst], pv);
}

// ---------------------------------------------------------------------------
// Kernel 5: out[dst] += (p/z[dst]) * x_l[src]. One wave per edge,
// float4 gather + 4 global_atomic_add_f32 per lane.
// ---------------------------------------------------------------------------
__global__ __launch_bounds__(256) void gat_scatter(
    const int* __restrict__ ei, const float* __restrict__ xl,
    const float* __restrict__ p, const float* __restrict__ z,
    float* __restrict__ out, int E, int ET)
{
  int wave = (blockIdx.x * blockDim.x + threadIdx.x) >> 5;
  int lane = threadIdx.x & 31;
  if (wave >= ET) return;
  int src, dst;
  if (wave < E) { src = ei[wave]; dst = ei[E + wave]; }
  else          { src = wave - E; dst = src; }

  float alpha = p[wave] / z[dst];
  float4 l = ((const float4*)(xl + (size_t)src * D))[lane];
  float* o = out + (size_t)dst * D + lane * 4;
  atomicAdd(o + 0, alpha * l.x);
  atomicAdd(o + 1, alpha * l.y);
  atomicAdd(o + 2, alpha * l.z);
  atomicAdd(o + 3, alpha * l.w);
}

// ---------------------------------------------------------------------------
// Kernel 6: out = leaky_relu(out + bias, 0.01), in place
// ---------------------------------------------------------------------------
__global__ void gat_final(float* __restrict__ out,
                          const float* __restrict__ bias, int N)
{
  int i = blockIdx.x * blockDim.x + threadIdx.x;
  if (i >= N * D) return;
  float v = out[i] + bias[i & (D - 1)];
  out[i] = v > 0.f ? v : NEG_OUT * v;
}

extern "C" void kernel_launch(void* const* d_in, const int* in_sizes, int n_in,
                              void* d_out, int out_size, void* d_ws, size_t ws_size,
                              hipStream_t stream)
{
  const float* x    = (const float*)d_in[0];
  const int*   ei   = (const int*)d_in[1];
  // d_in[2] = edge_attr: unused by reference (edge_dim=None)
  const float* Wl   = (const float*)d_in[3];
  const float* bl   = (const float*)d_in[4];
  const float* Wr   = (const float*)d_in[5];
  const float* br   = (const float*)d_in[6];
  const float* att  = (const float*)d_in[7];
  const float* bias = (const float*)d_in[8];

  const int N  = in_sizes[0] / D;
  const int E  = in_sizes[1] / 2;
  const int ET = E + N;              // edges + self loops
  float* out = (float*)d_out;

  // carve workspace
  char* ws = (char*)d_ws;
  float* xl   = (float*)ws; ws += (size_t)N * D * sizeof(float);
  float* xr   = (float*)ws; ws += (size_t)N * D * sizeof(float);
  float* elog = (float*)ws; ws += (size_t)ET * sizeof(float);
  float* pbuf = (float*)ws; ws += (size_t)ET * sizeof(float);
  float* mx   = (float*)ws; ws += (size_t)N * sizeof(float);
  float* z    = (float*)ws; ws += (size_t)N * sizeof(float);

  gat_gemm<<<(N + 15) / 16, 128, 0, stream>>>(x, Wl, bl, Wr, br, xl, xr, N);
  gat_init<<<(N * D + 255) / 256, 256, 0, stream>>>(out, mx, z, N);
  gat_logit<<<(ET * 32 + 255) / 256, 256, 0, stream>>>(ei, xl, xr, att, elog, mx, E, ET);
  gat_expsum<<<(ET + 255) / 256, 256, 0, stream>>>(ei, elog, mx, pbuf, z, E, ET);
  gat_scatter<<<(ET * 32 + 255) / 256, 256, 0, stream>>>(ei, xl, pbuf, z, out, E, ET);
  gat_final<<<(N * D + 255) / 256, 256, 0, stream>>>(out, bias, N);
}